// FeatTimeAttention_13752485281797
// MI455X (gfx1250) — compile-verified
//
#include <hip/hip_runtime.h>
#include <hip/hip_bf16.h>

// Problem constants (from reference): B=64, T=256, D_F=32, L=128
#define PB   64
#define PT   256
#define PDF  32
#define PL   128
#define TPB  8        // time steps per block
#define LP   132      // padded LDS row stride for the 32x128 feature matrix
#define AST  33       // padded stride for 32x32 matrices

typedef __attribute__((ext_vector_type(2))) float v2f;
typedef __attribute__((ext_vector_type(8))) float v8f;

// ---------------------------------------------------------------------------
// Prep kernel: compute sum(|beta|) into ws[0], zero the output accumulator.
// ---------------------------------------------------------------------------
__global__ void ftattn_prep_kernel(const float* __restrict__ beta,
                                   float* __restrict__ out,
                                   float* __restrict__ wsum) {
    __shared__ float red[PT];
    const int tid = threadIdx.x;            // 256 threads
    red[tid] = fabsf(beta[tid]);
    __syncthreads();
    for (int s = PT / 2; s > 0; s >>= 1) {
        if (tid < s) red[tid] += red[tid + s];
        __syncthreads();
    }
    if (tid == 0) wsum[0] = red[0];
    for (int i = tid; i < PB * PL; i += PT) out[i] = 0.0f;
}

// ---------------------------------------------------------------------------
// Main kernel: one block = (b, 8 consecutive t).  128 threads = 4 waves.
// ---------------------------------------------------------------------------
__global__ void __launch_bounds__(128)
ftattn_main_kernel(const float* __restrict__ x,       // [B,T,32]
                   const float* __restrict__ latent,  // [B,T,128]
                   const float* __restrict__ kern,    // [32,128]
                   const float* __restrict__ bias,    // [32,128]
                   const float* __restrict__ beta,    // [T]
                   const float* __restrict__ wsum,    // [1] = sum|beta|
                   float* __restrict__ out) {         // [B,128]
    __shared__ float sKer[PDF * LP];
    __shared__ float sBias[PDF * LP];
    __shared__ float sF[PDF * LP];       // features, row-major [32][132]
    __shared__ float sA[PDF * AST];      // XtX -> reduced to I
    __shared__ float sInv[PDF * AST];    // I -> (XtX)^-1
    __shared__ float sX[PDF];            // x row
    __shared__ float sLat[PL];           // latent row
    __shared__ float sXty[PDF];
    __shared__ float sScores[PDF];

    const int tid  = threadIdx.x;        // 0..127
    const int b    = blockIdx.y;
    const int t0   = blockIdx.x * TPB;

    // wave / lane decomposition for WMMA tiling of the 32x32 Gram matrix
    const int lane = tid & 31;
    const int wave = tid >> 5;           // 0..3
    const int m0   = (wave >> 1) * 16;   // row tile base
    const int n0   = (wave & 1) * 16;    // col tile base
    const int half = lane >> 4;          // 0: K=0,1  1: K=2,3
    const int lr   = lane & 15;

    // Gauss-Jordan ownership: row r, 8-column group g (64 aug. columns total)
    const int gr = tid >> 2;             // 0..31
    const int gg = tid & 3;              // 0..3

    // Stage kernel & bias to LDS once per block (reused 8x).
    for (int idx = tid; idx < PDF * PL; idx += 128) {
        const int d = idx >> 7, l = idx & 127;
        sKer[d * LP + l]  = kern[idx];
        sBias[d * LP + l] = bias[idx];
    }

    const float inv_bsum = 1.0f / wsum[0];
    float acc = 0.0f;                    // this thread owns output column l = tid

    for (int tt = 0; tt < TPB; ++tt) {
        const int t = t0 + tt;

        // ---- stage x row and latent row -------------------------------
        if (tid < PDF) sX[tid] = x[((size_t)b * PT + t) * PDF + tid];
        sLat[tid] = latent[((size_t)b * PT + t) * PL + tid];

        // prefetch next t's rows (gfx1250 global_prefetch_b8)
        if (tt + 1 < TPB) {
            __builtin_prefetch(&x[((size_t)b * PT + t + 1) * PDF + (tid & 31)], 0, 1);
            __builtin_prefetch(&latent[((size_t)b * PT + t + 1) * PL + tid], 0, 1);
        }
        __syncthreads();

        // ---- features: F[d][l] = relu(x[d]*K[d][l] + bias[d][l]) ------
        // thread owns column l = tid across all 32 rows
        {
            const int l = tid;
            #pragma unroll
            for (int d = 0; d < PDF; ++d) {
                float v = fmaf(sX[d], sKer[d * LP + l], sBias[d * LP + l]);
                sF[d * LP + l] = v > 0.0f ? v : 0.0f;
            }
        }
        __syncthreads();

        // ---- XtX = F * F^T via V_WMMA_F32_16X16X4_F32 -----------------
        // Each wave computes one 16x16 tile, K=128 in steps of 4.
        {
            v8f c = {};
            const float* rowA = &sF[(m0 + lr) * LP];  // A: rows of F
            const float* rowB = &sF[(n0 + lr) * LP];  // B = F^T: B[k][n] = F[n][k]
            #pragma unroll
            for (int k0 = 0; k0 < PL; k0 += 4) {
                v2f a, bb;
                a.x  = rowA[k0 + 2 * half];      // K = 0 (or 2)
                a.y  = rowA[k0 + 2 * half + 1];  // K = 1 (or 3)
                bb.x = rowB[k0 + 2 * half];
                bb.y = rowB[k0 + 2 * half + 1];
                c = __builtin_amdgcn_wmma_f32_16x16x4_f32(
                        false, a, false, bb, (short)0, c, false, false);
            }
            // C/D layout: lanes 0-15 -> M = m0+r, lanes 16-31 -> M = m0+8+r; N = n0+lr
            #pragma unroll
            for (int r = 0; r < 8; ++r) {
                sA[(m0 + half * 8 + r) * AST + (n0 + lr)] = c[r];
            }
        }
        // init identity for the inverse
        for (int idx = tid; idx < PDF * PDF; idx += 128) {
            const int rr = idx >> 5, cc = idx & 31;
            sInv[rr * AST + cc] = (rr == cc) ? 1.0f : 0.0f;
        }
        __syncthreads();

        // ---- Gauss-Jordan inverse of SPD 32x32 (no pivoting needed) ---
        for (int k = 0; k < PDF; ++k) {
            const float piv  = sA[k * AST + k];
            const float pinv = 1.0f / piv;
            __syncthreads();
            if (gr == k) {
                #pragma unroll
                for (int j = 0; j < 8; ++j) {
                    sA[k * AST + gg * 8 + j]   *= pinv;
                    sInv[k * AST + gg * 8 + j] *= pinv;
                }
            }
            __syncthreads();
            const float f = sA[gr * AST + k];
            if (gr != k) {
                #pragma unroll
                for (int j = 0; j < 8; ++j) {
                    const int cc = gg * 8 + j;
                    sA[gr * AST + cc]   -= f * sA[k * AST + cc];
                    sInv[gr * AST + cc] -= f * sInv[k * AST + cc];
                }
            }
            __syncthreads();
        }

        // ---- xty[d] = F[d][:] . latent -------------------------------
        if (tid < PDF) {
            float s = 0.0f;
            #pragma unroll 8
            for (int l = 0; l < PL; ++l) s = fmaf(sF[tid * LP + l], sLat[l], s);
            sXty[tid] = s;
        }
        __syncthreads();

        // ---- scores = XtX^-1 * xty -----------------------------------
        if (tid < PDF) {
            float s = 0.0f;
            #pragma unroll
            for (int e = 0; e < PDF; ++e) s = fmaf(sInv[tid * AST + e], sXty[e], s);
            sScores[tid] = s;
        }
        __syncthreads();

        // ---- o_hat[l] = scores . F[:,l]; weighted accumulate ----------
        {
            const int l = tid;
            float oh = 0.0f;
            #pragma unroll
            for (int d = 0; d < PDF; ++d) oh = fmaf(sScores[d], sF[d * LP + l], oh);
            acc = fmaf(fabsf(beta[t]) * inv_bsum, oh, acc);
        }
        __syncthreads();
    }

    atomicAdd(&out[b * PL + tid], acc);
}

// ---------------------------------------------------------------------------
extern "C" void kernel_launch(void* const* d_in, const int* in_sizes, int n_in,
                              void* d_out, int out_size, void* d_ws, size_t ws_size,
                              hipStream_t stream) {
    const float* x      = (const float*)d_in[0];  // [64,256,32]
    const float* latent = (const float*)d_in[1];  // [64,256,128]
    const float* kern   = (const float*)d_in[2];  // [1,1,32,128]
    const float* bias   = (const float*)d_in[3];  // [1,1,32,128]
    const float* beta   = (const float*)d_in[4];  // [1,256,1]
    float* out  = (float*)d_out;                  // [64,128]
    float* wsum = (float*)d_ws;

    ftattn_prep_kernel<<<1, PT, 0, stream>>>(beta, out, wsum);

    dim3 grid(PT / TPB, PB);
    ftattn_main_kernel<<<grid, 128, 0, stream>>>(x, latent, kern, bias, beta, wsum, out);
}